// ModelEVOLVE_75797582840082
// MI455X (gfx1250) — compile-verified
//
#include <hip/hip_runtime.h>
#include <hip/hip_bf16.h>
#include <math.h>

#define NN 50000
#define CC 256
#define EE 800000

typedef __attribute__((ext_vector_type(2))) float v2f;
typedef __attribute__((ext_vector_type(8))) float v8f;

__device__ __forceinline__ v8f wmma4(v2f a, v2f b, v8f c) {
    // D = A(16x4) * B(4x16) + C(16x16), fp32
    return __builtin_amdgcn_wmma_f32_16x16x4_f32(
        false, a, false, b, (short)0, c, false, false);
}

__device__ __forceinline__ unsigned f2ord(float f) {
    unsigned u = __float_as_uint(f);
    return (u & 0x80000000u) ? ~u : (u | 0x80000000u);
}

__device__ __forceinline__ float sigm(float x) { return 1.0f / (1.0f + expf(-x)); }

// ---------------- stage 1: ||p|| -------------------------------------------
__global__ void k_pnorm(const float* __restrict__ p, unsigned* __restrict__ meta) {
    __shared__ float red[256];
    int t = threadIdx.x;
    float v = p[t];
    red[t] = v * v;
    __syncthreads();
    for (int s = 128; s > 0; s >>= 1) {
        if (t < s) red[t] += red[t + s];
        __syncthreads();
    }
    if (t == 0) ((float*)meta)[2] = sqrtf(red[0]);
}

// ---------------- stage 2: scores = (x @ p) / ||p|| -------------------------
__global__ void k_score(const float* __restrict__ x, const float* __restrict__ p,
                        const unsigned* __restrict__ meta, float* __restrict__ scores) {
    int gid  = blockIdx.x * blockDim.x + threadIdx.x;
    int node = gid >> 5;
    int lane = threadIdx.x & 31;
    if (node >= NN) return;
    const float* row = x + (size_t)node * CC;
    float s = 0.f;
#pragma unroll
    for (int j = 0; j < 8; j++) s += row[lane + 32 * j] * p[lane + 32 * j];
    for (int m = 16; m >= 1; m >>= 1) s += __shfl_xor(s, m);
    if (lane == 0) scores[node] = s / ((const float*)meta)[2];
}

// ---------------- stage 3: 16-bit histogram --------------------------------
__global__ void k_hist(const float* __restrict__ scores, unsigned* __restrict__ hist) {
    int i = blockIdx.x * blockDim.x + threadIdx.x;
    if (i < NN) atomicAdd(&hist[f2ord(scores[i]) >> 16], 1u);
}

// ---------------- stage 4: find threshold bucket ---------------------------
__global__ void __launch_bounds__(1024) k_findbucket(const unsigned* __restrict__ hist,
                                                     unsigned* __restrict__ meta) {
    __shared__ unsigned chunk[1024];
    int t = threadIdx.x;
    unsigned s = 0;
    int base = 65536 - 64 * (t + 1);
#pragma unroll 4
    for (int i = 0; i < 64; i++) s += hist[base + i];
    chunk[t] = s;
    __syncthreads();
    if (t == 0) {
        unsigned cum = 0;
        int c = 0;
        for (; c < 1024; c++) {
            if (cum + chunk[c] >= 256u) break;
            cum += chunk[c];
        }
        int b = 65536 - 64 * c - 1;
        for (;; b--) {
            cum += hist[b];
            if (cum >= 256u || b == 0) break;
        }
        meta[0] = ((unsigned)b) << 16;   // threshold key
    }
}

// ---------------- stage 5: compact candidates ------------------------------
__global__ void k_compact(const float* __restrict__ scores, unsigned* __restrict__ meta,
                          float* __restrict__ cs, unsigned* __restrict__ ci) {
    int i = blockIdx.x * blockDim.x + threadIdx.x;
    if (i >= NN) return;
    float v = scores[i];
    if (f2ord(v) >= meta[0]) {
        unsigned pos = atomicAdd(&meta[1], 1u);
        if (pos < 2048u) { cs[pos] = v; ci[pos] = (unsigned)i; }
    }
}

// ---------------- stage 6: bitonic sort + x_tilde --------------------------
__global__ void __launch_bounds__(1024) k_topk(const float* __restrict__ cs,
                                               const unsigned* __restrict__ ci,
                                               const unsigned* __restrict__ meta,
                                               const float* __restrict__ x,
                                               float* __restrict__ x_tilde) {
    __shared__ float    ss[2048];
    __shared__ unsigned si[2048];
    int t = threadIdx.x;
    unsigned cnt = meta[1];
    if (cnt > 2048u) cnt = 2048u;
    for (int j = t; j < 2048; j += 1024) {
        if ((unsigned)j < cnt) { ss[j] = cs[j]; si[j] = ci[j]; }
        else                   { ss[j] = -INFINITY; si[j] = 0u; }
    }
    __syncthreads();
    // full bitonic sort, descending
    for (int k = 2; k <= 2048; k <<= 1) {
        for (int j = k >> 1; j > 0; j >>= 1) {
            for (int i = t; i < 2048; i += 1024) {
                int ixj = i ^ j;
                if (ixj > i) {
                    bool desc = ((i & k) == 0);
                    bool sw = desc ? (ss[i] < ss[ixj]) : (ss[i] > ss[ixj]);
                    if (sw) {
                        float tf = ss[i]; ss[i] = ss[ixj]; ss[ixj] = tf;
                        unsigned tu = si[i]; si[i] = si[ixj]; si[ixj] = tu;
                    }
                }
            }
            __syncthreads();
        }
    }
    // x_tilde[r][c] = x[perm[r]][c] * tanh(score[r]),  r < 256
    for (int it = 0; it < 64; it++) {
        int idx = t + 1024 * it;
        int r = idx >> 8, c = idx & 255;
        x_tilde[idx] = x[(size_t)si[r] * CC + c] * tanhf(ss[r]);
    }
}

// ---------------- stage 7: GRU gate GEMMs (WMMA f32 16x16x4) ---------------
__global__ void __launch_bounds__(32) k_gru(const float* __restrict__ x_tilde,
                                            const float* __restrict__ W_init,
                                            const float* __restrict__ W_ih,
                                            const float* __restrict__ W_hh,
                                            const float* __restrict__ b_ih,
                                            const float* __restrict__ b_hh,
                                            float* __restrict__ gi,
                                            float* __restrict__ gh) {
    int which = blockIdx.z;                       // 0 -> gi, 1 -> gh
    const float* A    = which ? W_init : x_tilde; // [256][256]
    const float* Wt   = which ? W_hh   : W_ih;    // [768][256]
    const float* bias = which ? b_hh   : b_ih;    // [768]
    float* out        = which ? gh     : gi;      // [256][768]
    int lane = threadIdx.x, h = lane >> 4, nl = lane & 15;
    int m0 = blockIdx.y * 16, n0 = blockIdx.x * 16;
    v8f acc = {0.f,0.f,0.f,0.f,0.f,0.f,0.f,0.f};
    for (int k0 = 0; k0 < 256; k0 += 4) {
        int kk = k0 + 2 * h;
        v2f a; a.x = A[(m0 + nl) * 256 + kk];  a.y = A[(m0 + nl) * 256 + kk + 1];
        v2f b; b.x = Wt[(n0 + nl) * 256 + kk]; b.y = Wt[(n0 + nl) * 256 + kk + 1];
        acc = wmma4(a, b, acc);
    }
    float bn = bias[n0 + nl];
#pragma unroll
    for (int r = 0; r < 8; r++)
        out[(m0 + r + 8 * h) * 768 + n0 + nl] = acc[r] + bn;
}

// ---------------- stage 8: gates -> evolved weight W -----------------------
__global__ void k_gates(const float* __restrict__ gi, const float* __restrict__ gh,
                        const float* __restrict__ W_init, float* __restrict__ W_evt) {
    int idx = blockIdx.x * 256 + threadIdx.x;     // 65536 elements
    int i = idx >> 8, j = idx & 255;
    float r = sigm(gi[i * 768 + j]       + gh[i * 768 + j]);
    float z = sigm(gi[i * 768 + 256 + j] + gh[i * 768 + 256 + j]);
    float n = tanhf(gi[i * 768 + 512 + j] + r * gh[i * 768 + 512 + j]);
    W_evt[idx] = (1.f - z) * n + z * W_init[idx];
}

// ---------------- stage 9: xw = x @ W (WMMA, 64 nodes/block) ---------------
// Dynamic LDS: A tile 64 x 257 floats (stride%64==1 -> conflict-free reads).
// B fragments loaded once per K-step, shared by 4 M-tile accumulator pairs.
__global__ void __launch_bounds__(256) k_xw(const float* __restrict__ x,
                                            const float* __restrict__ W,
                                            float* __restrict__ xw) {
    extern __shared__ float xs[];                 // [64][257]
    const int LD = 257;
    int tid = threadIdx.x;
    int node0 = blockIdx.x * 64;
    for (int it = 0; it < 64; it++) {
        int idx = tid + 256 * it;                 // 64*256 elements
        int r = idx >> 8, c = idx & 255;
        int row = node0 + r;
        xs[r * LD + c] = (row < NN) ? x[(size_t)row * CC + c] : 0.f;
    }
    __syncthreads();
    int wave = tid >> 5, lane = tid & 31, h = lane >> 4, nl = lane & 15;
    int n0a = wave * 16, n0b = (wave + 8) * 16;
    v8f acc[4][2];
#pragma unroll
    for (int mt = 0; mt < 4; mt++) {
        acc[mt][0] = (v8f){0.f,0.f,0.f,0.f,0.f,0.f,0.f,0.f};
        acc[mt][1] = (v8f){0.f,0.f,0.f,0.f,0.f,0.f,0.f,0.f};
    }
    for (int k0 = 0; k0 < 256; k0 += 4) {
        int kk = k0 + 2 * h;
        v2f b0; b0.x = W[kk * 256 + n0a + nl];  b0.y = W[(kk + 1) * 256 + n0a + nl];
        v2f b1; b1.x = W[kk * 256 + n0b + nl];  b1.y = W[(kk + 1) * 256 + n0b + nl];
#pragma unroll
        for (int mt = 0; mt < 4; mt++) {
            v2f a; a.x = xs[(mt * 16 + nl) * LD + kk];
                   a.y = xs[(mt * 16 + nl) * LD + kk + 1];
            acc[mt][0] = wmma4(a, b0, acc[mt][0]);
            acc[mt][1] = wmma4(a, b1, acc[mt][1]);
        }
    }
#pragma unroll
    for (int mt = 0; mt < 4; mt++) {
#pragma unroll
        for (int r = 0; r < 8; r++) {
            int row = node0 + mt * 16 + r + 8 * h;
            if (row < NN) {
                size_t base = (size_t)row * CC;
                xw[base + n0a + nl] = acc[mt][0][r];
                xw[base + n0b + nl] = acc[mt][1][r];
            }
        }
    }
}

// ---------------- stage 10: degree -----------------------------------------
__global__ void k_deg(const long long* __restrict__ eidx, const float* __restrict__ attr,
                      float* __restrict__ deg) {
    int e = blockIdx.x * blockDim.x + threadIdx.x;
    if (e < EE) atomicAdd(&deg[(int)eidx[EE + e]], attr[e]);  // dst = row 1
}

__global__ void k_dinv(const float* __restrict__ deg, float* __restrict__ dinv) {
    int i = blockIdx.x * blockDim.x + threadIdx.x;
    if (i < NN) { float d = deg[i]; dinv[i] = (d > 0.f) ? rsqrtf(d) : 0.f; }
}

// ---------------- stage 11: edge scatter-add -------------------------------
__global__ void k_scatter(const long long* __restrict__ eidx, const float* __restrict__ attr,
                          const float* __restrict__ dinv, const float* __restrict__ xw,
                          float* __restrict__ node_emb) {
    int gid  = blockIdx.x * blockDim.x + threadIdx.x;
    int e    = gid >> 5;
    int lane = threadIdx.x & 31;
    if (e >= EE) return;
    int s = (int)eidx[e], d = (int)eidx[EE + e];
    float nrm = dinv[s] * attr[e] * dinv[d];
    const float* xr = xw + (size_t)s * CC;
    float* er = node_emb + (size_t)d * CC;
#pragma unroll
    for (int j = 0; j < 8; j++) {
        int c = lane + 32 * j;
        atomicAdd(&er[c], xr[c] * nrm);
    }
}

// ---------------- stage 12: fused edge decoder (WMMA, 64 edges/block) ------
// Dynamic LDS: emb tile 64 x 513 floats + index/accumulator arrays.
// B fragments (lin1_w) loaded once per K-step, shared by 4 M-tiles: cuts
// lin1_w L2 traffic 4x versus 16-edge blocking.
__global__ void __launch_bounds__(256) k_edge(const long long* __restrict__ ewidx,
                                              const float* __restrict__ node_emb,
                                              const float* __restrict__ lin1_w,
                                              const float* __restrict__ lin1_b,
                                              const float* __restrict__ lin2_w,
                                              const float* __restrict__ lin2_b,
                                              float* __restrict__ pred) {
    extern __shared__ char smem[];
    const int LD = 513;
    float* emb     = (float*)smem;                           // [64][513]
    int*   sA      = (int*)(smem + 64 * LD * sizeof(float)); // [64]
    int*   tA      = sA + 64;                                // [64]
    float* predAcc = (float*)(tA + 64);                      // [64]

    int tid = threadIdx.x;
    int e0 = blockIdx.x * 64;
    if (tid < 64) {
        sA[tid] = (int)ewidx[e0 + tid];
        tA[tid] = (int)ewidx[EE + e0 + tid];
        predAcc[tid] = 0.f;
    }
    __syncthreads();
    for (int it = 0; it < 128; it++) {
        int idx = tid + 256 * it;                 // 64*512 elements
        int r = idx >> 9, c = idx & 511;
        int row = (c < 256) ? sA[r] : tA[r];
        emb[r * LD + c] = node_emb[(size_t)row * CC + (c & 255)];
    }
    __syncthreads();

    int wave = tid >> 5, lane = tid & 31, h = lane >> 4, nl = lane & 15;
    int oa = wave * 16 + nl, ob = (wave + 8) * 16 + nl;
    v8f acc[4][2];
#pragma unroll
    for (int mt = 0; mt < 4; mt++) {
        acc[mt][0] = (v8f){0.f,0.f,0.f,0.f,0.f,0.f,0.f,0.f};
        acc[mt][1] = (v8f){0.f,0.f,0.f,0.f,0.f,0.f,0.f,0.f};
    }
    for (int k0 = 0; k0 < 512; k0 += 4) {
        int kk = k0 + 2 * h;
        v2f b0; b0.x = lin1_w[(size_t)oa * 512 + kk]; b0.y = lin1_w[(size_t)oa * 512 + kk + 1];
        v2f b1; b1.x = lin1_w[(size_t)ob * 512 + kk]; b1.y = lin1_w[(size_t)ob * 512 + kk + 1];
#pragma unroll
        for (int mt = 0; mt < 4; mt++) {
            v2f a; a.x = emb[(mt * 16 + nl) * LD + kk];
                   a.y = emb[(mt * 16 + nl) * LD + kk + 1];
            acc[mt][0] = wmma4(a, b0, acc[mt][0]);
            acc[mt][1] = wmma4(a, b1, acc[mt][1]);
        }
    }
    float ba = lin1_b[oa], wa = lin2_w[oa];
    float bb = lin1_b[ob], wb = lin2_w[ob];
#pragma unroll
    for (int mt = 0; mt < 4; mt++) {
#pragma unroll
        for (int r = 0; r < 8; r++) {
            float h0 = acc[mt][0][r] + ba; h0 = h0 > 0.f ? h0 : 0.f;
            float h1 = acc[mt][1][r] + bb; h1 = h1 > 0.f ? h1 : 0.f;
            float v = h0 * wa + h1 * wb;
            for (int m = 8; m >= 1; m >>= 1) v += __shfl_xor(v, m);  // 16 n-lanes
            if (nl == 0) atomicAdd(&predAcc[mt * 16 + r + 8 * h], v);
        }
    }
    __syncthreads();
    if (tid < 64) pred[e0 + tid] = predAcc[tid] + lin2_b[0];
}

// ---------------------------------------------------------------------------
extern "C" void kernel_launch(void* const* d_in, const int* in_sizes, int n_in,
                              void* d_out, int out_size, void* d_ws, size_t ws_size,
                              hipStream_t stream) {
    const float*     x       = (const float*)d_in[0];
    const long long* eidx    = (const long long*)d_in[1];
    const float*     eattr   = (const float*)d_in[2];
    const long long* ewidx   = (const long long*)d_in[3];
    const float*     p_pool  = (const float*)d_in[4];
    const float*     W_ih    = (const float*)d_in[5];
    const float*     W_hh    = (const float*)d_in[6];
    const float*     b_ih    = (const float*)d_in[7];
    const float*     b_hh    = (const float*)d_in[8];
    const float*     W_init  = (const float*)d_in[9];
    const float*     lin1_w  = (const float*)d_in[10];
    const float*     lin1_b  = (const float*)d_in[11];
    const float*     lin2_w  = (const float*)d_in[12];
    const float*     lin2_b  = (const float*)d_in[13];
    float* pred = (float*)d_out;

    char* ws = (char*)d_ws;
    size_t off = 0;
    auto take = [&](size_t bytes) -> char* {
        char* ptr = ws + off;
        off = (off + bytes + 255) & ~(size_t)255;
        return ptr;
    };
    float*    scores   = (float*)take((size_t)NN * 4);
    unsigned* hist     = (unsigned*)take(65536 * 4);
    unsigned* meta     = (unsigned*)take(64);
    float*    cand_s   = (float*)take(2048 * 4);
    unsigned* cand_i   = (unsigned*)take(2048 * 4);
    float*    x_tilde  = (float*)take(256 * 256 * 4);
    float*    gi       = (float*)take(256 * 768 * 4);
    float*    gh       = (float*)take(256 * 768 * 4);
    float*    W_evt    = (float*)take(256 * 256 * 4);
    float*    deg      = (float*)take((size_t)NN * 4);
    float*    dinv     = (float*)take((size_t)NN * 4);
    float*    xw       = (float*)take((size_t)NN * CC * 4);
    float*    node_emb = (float*)take((size_t)NN * CC * 4);

    hipMemsetAsync(hist, 0, 65536 * 4, stream);
    hipMemsetAsync(meta, 0, 64, stream);
    hipMemsetAsync(deg, 0, (size_t)NN * 4, stream);
    hipMemsetAsync(node_emb, 0, (size_t)NN * CC * 4, stream);

    k_pnorm<<<1, 256, 0, stream>>>(p_pool, meta);
    k_score<<<(NN * 32 + 255) / 256, 256, 0, stream>>>(x, p_pool, meta, scores);
    k_hist<<<(NN + 255) / 256, 256, 0, stream>>>(scores, hist);
    k_findbucket<<<1, 1024, 0, stream>>>(hist, meta);
    k_compact<<<(NN + 255) / 256, 256, 0, stream>>>(scores, meta, cand_s, cand_i);
    k_topk<<<1, 1024, 0, stream>>>(cand_s, cand_i, meta, x, x_tilde);
    k_gru<<<dim3(48, 16, 2), 32, 0, stream>>>(x_tilde, W_init, W_ih, W_hh, b_ih, b_hh, gi, gh);
    k_gates<<<256, 256, 0, stream>>>(gi, gh, W_init, W_evt);

    size_t xw_lds = (size_t)64 * 257 * sizeof(float);                 // ~66 KB
    k_xw<<<(NN + 63) / 64, 256, xw_lds, stream>>>(x, W_evt, xw);

    k_deg<<<(EE + 255) / 256, 256, 0, stream>>>(eidx, eattr, deg);
    k_dinv<<<(NN + 255) / 256, 256, 0, stream>>>(deg, dinv);
    k_scatter<<<(EE * 32 + 255) / 256, 256, 0, stream>>>(eidx, eattr, dinv, xw, node_emb);

    size_t edge_lds = (size_t)64 * 513 * sizeof(float) + 3 * 64 * 4;  // ~132 KB
    k_edge<<<EE / 64, 256, edge_lds, stream>>>(ewidx, node_emb, lin1_w, lin1_b, lin2_w,
                                               lin2_b, pred);
}